// SoftFlattenLoss_62929860821311
// MI455X (gfx1250) — compile-verified
//
#include <hip/hip_runtime.h>
#include <hip/hip_bf16.h>

#define EPS 1e-6f
#define NTHR 256
#define NBLK 4096

struct F3 { float x, y, z; };

__device__ __forceinline__ float dot3(const F3& a, const F3& b) {
    return fmaf(a.x, b.x, fmaf(a.y, b.y, a.z * b.z));
}

// Stage 1: per-edge math + per-block deterministic partial sum.
__global__ __launch_bounds__(NTHR) void soft_flatten_edge_kernel(
    const float* __restrict__ verts,
    const int*   __restrict__ v0s,
    const int*   __restrict__ v1s,
    const int*   __restrict__ v2s,
    const int*   __restrict__ v3s,
    float*       __restrict__ cos_out,   // d_out + 1, E elements
    float*       __restrict__ partials,  // d_ws, NBLK floats
    int E)
{
    const int stride = NBLK * NTHR;
    const F3* __restrict__ vv = (const F3*)verts;

    float acc = 0.0f;
    for (int e = blockIdx.x * NTHR + threadIdx.x; e < E; e += stride) {
        // Software-pipeline the streamed index arrays one grid-stride ahead
        // (lowers to global_prefetch_b8 on gfx1250).
        int en = e + stride;
        if (en < E) {
            __builtin_prefetch(&v0s[en], 0, 0);
            __builtin_prefetch(&v1s[en], 0, 0);
            __builtin_prefetch(&v2s[en], 0, 0);
            __builtin_prefetch(&v3s[en], 0, 0);
        }

        int i0 = v0s[e], i1 = v1s[e], i2 = v2s[e], i3 = v3s[e];
        // 12B gathers; vertices (24 MB) are L2-resident on MI455X (192 MB L2).
        F3 p0 = vv[i0];
        F3 p1 = vv[i1];
        F3 p2 = vv[i2];
        F3 p3 = vv[i3];

        F3 a  = { p1.x - p0.x, p1.y - p0.y, p1.z - p0.z };
        F3 b1 = { p2.x - p0.x, p2.y - p0.y, p2.z - p0.z };
        F3 b2 = { p3.x - p0.x, p3.y - p0.y, p3.z - p0.z };

        float al2     = dot3(a, a);
        float al1     = sqrtf(al2 + EPS);
        float inv_al2 = 1.0f / (al2 + EPS);

        // wing(a, b1)
        float b1l2 = dot3(b1, b1);
        float b1l1 = sqrtf(b1l2 + EPS);
        float ab1  = dot3(a, b1);
        float c1   = ab1 / (al1 * b1l1 + EPS);
        float s1   = sqrtf(1.0f - c1 * c1 + EPS);
        float t1   = ab1 * inv_al2;
        F3 cb1 = { b1.x - a.x * t1, b1.y - a.y * t1, b1.z - a.z * t1 };
        float cb1l1 = b1l1 * s1;

        // wing(a, b2)
        float b2l2 = dot3(b2, b2);
        float b2l1 = sqrtf(b2l2 + EPS);
        float ab2  = dot3(a, b2);
        float c2   = ab2 / (al1 * b2l1 + EPS);
        float s2   = sqrtf(1.0f - c2 * c2 + EPS);
        float t2   = ab2 * inv_al2;
        F3 cb2 = { b2.x - a.x * t2, b2.y - a.y * t2, b2.z - a.z * t2 };
        float cb2l1 = b2l1 * s2;

        float c = dot3(cb1, cb2) / (cb1l1 * cb2l1 + EPS);

        cos_out[e] = c;                 // cos_ori (stop_gradient copy)
        float cp1 = c + 1.0f;
        acc = fmaf(cp1, cp1, acc);      // fixed per-thread order: deterministic
    }

    // wave32 butterfly reduce (gfx1250 is wave32 — do NOT assume 64 lanes)
    #pragma unroll
    for (int off = 16; off > 0; off >>= 1)
        acc += __shfl_xor(acc, off, 32);

    __shared__ float smem[NTHR / 32];
    const int lane = threadIdx.x & 31;
    const int wave = threadIdx.x >> 5;
    if (lane == 0) smem[wave] = acc;
    __syncthreads();

    if (wave == 0) {
        float v = (lane < (NTHR / 32)) ? smem[lane] : 0.0f;
        #pragma unroll
        for (int off = 4; off > 0; off >>= 1)
            v += __shfl_xor(v, off, 32);
        if (lane == 0) partials[blockIdx.x] = v;
    }
}

// Stage 2: deterministic single-block reduction of the per-block partials.
__global__ __launch_bounds__(NTHR) void reduce_partials_kernel(
    const float* __restrict__ partials,
    float*       __restrict__ loss_out,  // d_out[0]
    int n)
{
    float acc = 0.0f;
    for (int i = threadIdx.x; i < n; i += NTHR)  // fixed order per thread
        acc += partials[i];

    #pragma unroll
    for (int off = 16; off > 0; off >>= 1)
        acc += __shfl_xor(acc, off, 32);

    __shared__ float smem[NTHR / 32];
    const int lane = threadIdx.x & 31;
    const int wave = threadIdx.x >> 5;
    if (lane == 0) smem[wave] = acc;
    __syncthreads();

    if (wave == 0) {
        float v = (lane < (NTHR / 32)) ? smem[lane] : 0.0f;
        #pragma unroll
        for (int off = 4; off > 0; off >>= 1)
            v += __shfl_xor(v, off, 32);
        if (lane == 0) loss_out[0] = v;
    }
}

extern "C" void kernel_launch(void* const* d_in, const int* in_sizes, int n_in,
                              void* d_out, int out_size, void* d_ws, size_t ws_size,
                              hipStream_t stream) {
    const float* verts = (const float*)d_in[0];
    const int*   v0s   = (const int*)d_in[1];
    const int*   v1s   = (const int*)d_in[2];
    const int*   v2s   = (const int*)d_in[3];
    const int*   v3s   = (const int*)d_in[4];
    const int    E     = in_sizes[1];

    float* out      = (float*)d_out;     // out[0] = loss, out[1..E] = cos_ori
    float* partials = (float*)d_ws;      // NBLK floats (16 KB)

    soft_flatten_edge_kernel<<<NBLK, NTHR, 0, stream>>>(
        verts, v0s, v1s, v2s, v3s, out + 1, partials, E);
    reduce_partials_kernel<<<1, NTHR, 0, stream>>>(partials, out, NBLK);
}